// update_v_6975026889058
// MI455X (gfx1250) — compile-verified
//
#include <hip/hip_runtime.h>
#include <hip/hip_bf16.h>

typedef __attribute__((ext_vector_type(2))) float v2f;
typedef __attribute__((ext_vector_type(8))) float v8f;

#define H 128          // hidden / filter width
#define CCAT 256       // concat width

__device__ __forceinline__ float ssp(float x) {
    // softplus(x) - log(2), numerically stable
    float sp = fmaxf(x, 0.0f) + log1pf(expf(-fabsf(x)));
    return sp - 0.69314718055994530942f;
}

__device__ __forceinline__ v8f wmma4(v2f a, v2f b, v8f c) {
    // D = A(16x4 f32) x B(4x16 f32) + C(16x16 f32)
    return __builtin_amdgcn_wmma_f32_16x16x4_f32(
        false, a, false, b, (short)0, c, false, false);
}

// One 16x16 output tile of A[16xK] * W^T (W is [128][K] row-major), bias-pre-loaded.
__device__ __forceinline__ v8f gemm_tile(const float* __restrict__ A, int lda,
                                         const float* __restrict__ W, int K,
                                         float biasv, int n0, int lane) {
    const int mr = lane & 15;
    const int kh = (lane >> 4) * 2;
    v8f acc;
#pragma unroll
    for (int r = 0; r < 8; ++r) acc[r] = biasv;
    for (int k = 0; k < K; k += 4) {
        v2f av, bv;
        const float* ap = A + mr * lda + k + kh;
        av.x = ap[0];
        av.y = ap[1];
        const float* wp = W + (n0 + mr) * K + k + kh;   // B[k][n] = W[n][k]
        bv.x = wp[0];
        bv.y = wp[1];
        acc = wmma4(av, bv, acc);
    }
    return acc;
}

// ---------------- zero workspace ----------------
__global__ void zero_f4_kernel(float4* __restrict__ p, long long n4) {
    long long gid = (long long)blockIdx.x * blockDim.x + threadIdx.x;
    if (gid < n4) p[gid] = make_float4(0.f, 0.f, 0.f, 0.f);
}

// ---------------- scatter-add: acc[idx[e]] += e[e,:] ----------------
// one thread = one edge x 4 features (32 threads span the 128-wide row)
__global__ void scatter_add_kernel(const float* __restrict__ efeat,
                                   const int* __restrict__ idx_row1,
                                   float* __restrict__ acc,
                                   long long num_edges) {
    long long gid = (long long)blockIdx.x * blockDim.x + threadIdx.x;
    long long edge = gid >> 5;
    int f = (int)(gid & 31) * 4;
    if (edge >= num_edges) return;
    long long dst = (long long)idx_row1[edge];
    float4 val = *(const float4*)(efeat + edge * H + f);
    float* p = acc + dst * H + f;
    atomicAdd(p + 0, val.x);
    atomicAdd(p + 1, val.y);
    atomicAdd(p + 2, val.z);
    atomicAdd(p + 3, val.w);
}

// ---------------- fused MLP: one wave per 16-node tile ----------------
__global__ __launch_bounds__(32)
void fused_mlp_kernel(const float* __restrict__ v,
                      const float* __restrict__ agg1,
                      const float* __restrict__ agg2,
                      const float* __restrict__ W1,  const float* __restrict__ b1,
                      const float* __restrict__ W2,  const float* __restrict__ b2,
                      const float* __restrict__ W1h, const float* __restrict__ b1h,
                      const float* __restrict__ W2h, const float* __restrict__ b2h,
                      const float* __restrict__ Wc,  const float* __restrict__ bc,
                      float* __restrict__ out, int n_nodes) {
    __shared__ float sh_h[16 * H];       // stage-1 activations (A-layout source)
    __shared__ float sh_cat[16 * CCAT];  // concat buffer

    const int lane = threadIdx.x;
    const int mr = lane & 15;
    const int mh = (lane >> 4) * 8;      // C/D row offset for this half-wave
    const long long row0 = (long long)blockIdx.x * 16;

    // two branches: (agg, W1, b1, W2, b2) -> sh_cat[:, 0:128], hull -> [:,128:256]
    for (int br = 0; br < 2; ++br) {
        const float* agg = br ? agg2 : agg1;
        const float* Wa  = br ? W1h : W1;
        const float* ba  = br ? b1h : b1;
        const float* Wb  = br ? W2h : W2;
        const float* bb  = br ? b2h : b2;
        const int cat0   = br ? H : 0;

        // stage 1: h = ssp(agg @ Wa^T + ba) -> LDS
        {
            // clamp row for safe load (store side is exact-guarded below)
            long long r = row0 + mr;
            if (r >= n_nodes) r = n_nodes - 1;
            const float* A = agg + r * H - (long long)mr * H;  // base so A[mr*H + ..] = agg[r][..]
            for (int nt = 0; nt < 8; ++nt) {
                int n0 = nt * 16;
                v8f acc = gemm_tile(A, H, Wa, H, ba[n0 + mr], n0, lane);
#pragma unroll
                for (int rr = 0; rr < 8; ++rr)
                    sh_h[(rr + mh) * H + n0 + mr] = ssp(acc[rr]);
            }
        }
        __syncthreads();

        // stage 2: t = h @ Wb^T + bb -> sh_cat[:, cat0:cat0+128]
        for (int nt = 0; nt < 8; ++nt) {
            int n0 = nt * 16;
            v8f acc = gemm_tile(sh_h, H, Wb, H, bb[n0 + mr], n0, lane);
#pragma unroll
            for (int rr = 0; rr < 8; ++rr)
                sh_cat[(rr + mh) * CCAT + cat0 + n0 + mr] = acc[rr];
        }
        __syncthreads();
    }

    // stage 3: out = v + ssp(cat @ Wc^T + bc), Wc is [128][256]
    for (int nt = 0; nt < 8; ++nt) {
        int n0 = nt * 16;
        v8f acc = gemm_tile(sh_cat, CCAT, Wc, CCAT, bc[n0 + mr], n0, lane);
#pragma unroll
        for (int rr = 0; rr < 8; ++rr) {
            long long node = row0 + rr + mh;
            if (node < n_nodes) {
                long long o = node * H + n0 + mr;
                out[o] = v[o] + ssp(acc[rr]);
            }
        }
    }
}

extern "C" void kernel_launch(void* const* d_in, const int* in_sizes, int n_in,
                              void* d_out, int out_size, void* d_ws, size_t ws_size,
                              hipStream_t stream) {
    const float* v    = (const float*)d_in[0];
    const float* e    = (const float*)d_in[1];
    const int*  eidx  = (const int*)d_in[2];    // [2,E], row-major
    const float* eh   = (const float*)d_in[3];
    const int*  ehidx = (const int*)d_in[4];    // [2,Eh]
    const float* W1   = (const float*)d_in[5];
    const float* b1   = (const float*)d_in[6];
    const float* W2   = (const float*)d_in[7];
    const float* b2   = (const float*)d_in[8];
    const float* W1h  = (const float*)d_in[9];
    const float* b1h  = (const float*)d_in[10];
    const float* W2h  = (const float*)d_in[11];
    const float* b2h  = (const float*)d_in[12];
    const float* Wc   = (const float*)d_in[13];
    const float* bc   = (const float*)d_in[14];

    const long long N  = (long long)in_sizes[0] / H;
    const long long E  = (long long)in_sizes[1] / H;
    const long long Eh = (long long)in_sizes[3] / H;

    float* acc1 = (float*)d_ws;            // [N,128]
    float* acc2 = acc1 + N * H;            // [N,128]

    // zero accumulators (deterministic each call)
    long long n4 = (2 * N * H) / 4;
    {
        long long blocks = (n4 + 255) / 256;
        zero_f4_kernel<<<(unsigned)blocks, 256, 0, stream>>>((float4*)d_ws, n4);
    }

    // scatter-sum both edge sets (idx row 1 = destination node)
    {
        long long threads = E * 32;
        long long blocks = (threads + 255) / 256;
        scatter_add_kernel<<<(unsigned)blocks, 256, 0, stream>>>(e, eidx + E, acc1, E);
    }
    {
        long long threads = Eh * 32;
        long long blocks = (threads + 255) / 256;
        scatter_add_kernel<<<(unsigned)blocks, 256, 0, stream>>>(eh, ehidx + Eh, acc2, Eh);
    }

    // fused per-tile MLP with fp32 WMMA
    {
        long long blocks = (N + 15) / 16;
        fused_mlp_kernel<<<(unsigned)blocks, 32, 0, stream>>>(
            v, acc1, acc2, W1, b1, W2, b2, W1h, b1h, W2h, b2h, Wc, bc,
            (float*)d_out, (int)N);
    }
}